// YoloV1Loss_32366873542892
// MI455X (gfx1250) — compile-verified
//
#include <hip/hip_runtime.h>

#define NROWS 802816   // 16384 * 49
#define NBLK  3136     // NROWS / 256
#define TPB   256
#define SCAN_CHUNK 13  // 256 * 13 = 3328 >= 3136

typedef float v2f __attribute__((ext_vector_type(2)));
typedef float v8f __attribute__((ext_vector_type(8)));

// Exact f32 wave32 reduction using V_WMMA_F32_16X16X4_F32.
// A: lane l<16 -> (M=l, K=0), lane l>=16 -> (M=l-16, K=2); other K slots zero.
// B: all ones => D[m][n] = x_m + x_{m+16}. Summing D's 8 VGPRs per lane gives
// lanes 0-15: sum over M=0..7, lanes 16-31: sum over M=8..15. Combine lane 0 + lane 16.
// Must be called with EXEC all ones (unconditionally, full waves).
__device__ __forceinline__ float wave_reduce_wmma(float x) {
  v2f a; a[0] = x;   a[1] = 0.0f;
  v2f b; b[0] = 1.0f; b[1] = 1.0f;
  v8f c = {};
  v8f d = __builtin_amdgcn_wmma_f32_16x16x4_f32(
      /*neg_a=*/false, a, /*neg_b=*/false, b,
      /*c_mod=*/(short)0, c, /*reuse_a=*/false, /*reuse_b=*/false);
  float s = ((d[0] + d[1]) + (d[2] + d[3])) + ((d[4] + d[5]) + (d[6] + d[7]));
  return __shfl(s, 0, 32) + __shfl(s, 16, 32);
}

// Per-row YOLO loss contribution + mask. Rows are 30 floats = 120 bytes,
// always 8-byte aligned -> float2 loads.
__device__ __forceinline__ float row_contrib(const float* __restrict__ p,
                                             const float* __restrict__ t,
                                             size_t row, int* mask_out) {
  const float2* pv = reinterpret_cast<const float2*>(p + row * 30); // p[0..9]
  const float2* tv = reinterpret_cast<const float2*>(t + row * 30); // t[0..7]
  float2 pq0 = pv[0], pq1 = pv[1], pq2 = pv[2], pq3 = pv[3], pq4 = pv[4];
  float2 tq0 = tv[0], tq1 = tv[1], tq2 = tv[2], tq3 = tv[3];
  float t8 = (t + row * 30)[8];

  float p0 = pq0.x, p1 = pq0.y, p2 = pq1.x, p3 = pq1.y;
  float p5 = pq2.y, p6 = pq3.x, p7 = pq3.y, p8 = pq4.x, p9 = pq4.y;
  float t0 = tq0.x, t1 = tq0.y, t2 = tq1.x, t3 = tq1.y;
  float t4 = tq2.x, t5 = tq2.y, t6 = tq3.x, t7 = tq3.y;

  *mask_out = (t4 == 1.0f) ? 1 : 0;

  // IoU(b0 = p0..p3, tb = t0..t3)
  float lt0 = fmaxf(p0, t0), lt1 = fmaxf(p1, t1);
  float rb0 = fminf(p2, t2), rb1 = fminf(p3, t3);
  float wh0 = fmaxf(rb0 - lt0, 0.0f), wh1 = fmaxf(rb1 - lt1, 0.0f);
  float inter0 = wh0 * wh1;
  float areaA0 = (p2 - p0) * (p3 - p1);
  float areaT  = (t2 - t0) * (t3 - t1);
  float i0 = inter0 / (areaA0 + areaT - inter0);

  // IoU(b1 = p5..p8, tb)
  float lu0 = fmaxf(p5, t0), lu1 = fmaxf(p6, t1);
  float ru0 = fminf(p7, t2), ru1 = fminf(p8, t3);
  float wj0 = fmaxf(ru0 - lu0, 0.0f), wj1 = fmaxf(ru1 - lu1, 0.0f);
  float inter1 = wj0 * wj1;
  float areaA1 = (p7 - p5) * (p8 - p6);
  float i1 = inter1 / (areaA1 + areaT - inter1);

  bool choose1 = (i1 > i0);
  float max_iou = choose1 ? i1 : i0;

  float pc0 = choose1 ? p5 : p0, pc1 = choose1 ? p6 : p1;
  float pc2 = choose1 ? p7 : p2, pc3 = choose1 ? p8 : p3;
  float tc0 = choose1 ? t5 : t0, tc1 = choose1 ? t6 : t1;
  float tc2 = choose1 ? t7 : t2, tc3 = choose1 ? t8 : t3;
  float conf = choose1 ? p9 : p5;

  float dx = pc0 - tc0, dy = pc1 - tc1;
  float dw = pc2 - tc2, dh = pc3 - tc3;
  float dc = conf - max_iou;
  return (dx * dx + dy * dy) + (dw * dw + dh * dh) + 2.0f * (dc * dc);
}

// In-block inclusive scan of mask bit: returns exclusive prefix for this thread,
// per-wave totals in wsum[8] (caller must have __shared__ int wsum[8]).
__device__ __forceinline__ int block_mask_scan(int m, int lane, int wave,
                                               int* wsum, int* wave_base_out) {
  int incl = m;
  #pragma unroll
  for (int d = 1; d < 32; d <<= 1) {
    int y = __shfl_up(incl, d, 32);
    if (lane >= d) incl += y;
  }
  int excl = incl - m;
  if (lane == 31) wsum[wave] = incl;
  __syncthreads();
  int base = 0;
  for (int i = 0; i < wave; ++i) base += wsum[i];
  *wave_base_out = base;
  return excl;
}

// ---------------- Fast path (needs ~3.25 MB scratch) ----------------

// Single streaming pass: contrib + mask, rank-ordered compaction per block.
__global__ void yolo_fuse(const float* __restrict__ p, const float* __restrict__ t,
                          float* __restrict__ cpack, int* __restrict__ counts) {
  int tid  = threadIdx.x;
  int lane = tid & 31, wave = tid >> 5;
  size_t row = (size_t)blockIdx.x * TPB + tid;

  int m;
  float contrib = row_contrib(p, t, row, &m);

  __shared__ int wsum[8];
  int base;
  int excl = block_mask_scan(m, lane, wave, wsum, &base);

  if (m) cpack[(size_t)blockIdx.x * TPB + base + excl] = contrib;

  if (tid == 0) {
    int tot = 0;
    for (int i = 0; i < 8; ++i) tot += wsum[i];
    counts[blockIdx.x] = tot;
  }
}

// Weighted sum over compacted contribs: entry i in block b is globally ranked
// offs[b]+i+1, so it counts iff i < half - offs[b] (and i < counts[b]).
__global__ void yolo_wsum(const float* __restrict__ cpack, const int* __restrict__ counts,
                          const int* __restrict__ offs, const int* __restrict__ scal,
                          float* __restrict__ partials) {
  int tid  = threadIdx.x;
  int lane = tid & 31, wave = tid >> 5;
  int b = blockIdx.x;

  int half  = scal[1];
  int limit = half - offs[b];
  int c     = counts[b];
  if (limit > c) limit = c;

  float v   = cpack[(size_t)b * TPB + tid];   // safe: within scratch
  float acc = (tid < limit) ? v : 0.0f;       // limit may be <= 0 -> all zero

  float wtot = wave_reduce_wmma(acc);
  __shared__ float fsum[8];
  if (lane == 0) fsum[wave] = wtot;
  __syncthreads();
  if (tid == 0) {
    float s = 0.0f;
    for (int i = 0; i < 8; ++i) s += fsum[i];
    partials[b] = s;
  }
}

// ---------------- Fallback path (small scratch, ~40 KB) ----------------

__global__ void yolo_count(const float* __restrict__ t, int* __restrict__ counts) {
  __shared__ int cnt;
  if (threadIdx.x == 0) cnt = 0;
  __syncthreads();
  size_t row = (size_t)blockIdx.x * TPB + threadIdx.x;
  int m = (t[row * 30 + 4] == 1.0f) ? 1 : 0;
  atomicAdd(&cnt, m);
  __syncthreads();
  if (threadIdx.x == 0) counts[blockIdx.x] = cnt;
}

__global__ void yolo_main(const float* __restrict__ p, const float* __restrict__ t,
                          const int* __restrict__ offs, const int* __restrict__ scal,
                          float* __restrict__ partials) {
  int tid  = threadIdx.x;
  int lane = tid & 31, wave = tid >> 5;
  size_t row = (size_t)blockIdx.x * TPB + tid;

  int m;
  float contrib = row_contrib(p, t, row, &m);

  __shared__ int wsum[8];
  int base;
  int excl = block_mask_scan(m, lane, wave, wsum, &base);

  int pre  = offs[blockIdx.x] + base + excl;  // masked rows strictly before this row
  int half = scal[1];
  float acc = (m && (pre < half)) ? contrib : 0.0f;

  float wtot = wave_reduce_wmma(acc);
  __shared__ float fsum[8];
  if (lane == 0) fsum[wave] = wtot;
  __syncthreads();
  if (tid == 0) {
    float s = 0.0f;
    for (int i = 0; i < 8; ++i) s += fsum[i];
    partials[blockIdx.x] = s;
  }
}

// ---------------- Shared passes ----------------

// Single-block exclusive scan of the 3136 block counts.
// scal[0] = n (total masked rows), scal[1] = n/2.
__global__ void yolo_scan(const int* __restrict__ counts, int* __restrict__ offs,
                          int* __restrict__ scal) {
  int tid  = threadIdx.x;
  int lane = tid & 31, wave = tid >> 5;
  int begin = tid * SCAN_CHUNK;
  int end   = begin + SCAN_CHUNK; if (end > NBLK) end = NBLK;
  int s = 0;
  for (int i = begin; i < end; ++i) s += counts[i];
  int incl = s;
  #pragma unroll
  for (int d = 1; d < 32; d <<= 1) {
    int y = __shfl_up(incl, d, 32);
    if (lane >= d) incl += y;
  }
  int excl = incl - s;
  __shared__ int wsum[8];
  if (lane == 31) wsum[wave] = incl;
  __syncthreads();
  int base = 0;
  for (int i = 0; i < wave; ++i) base += wsum[i];
  int run = base + excl;
  for (int i = begin; i < end; ++i) {
    offs[i] = run;
    run += counts[i];
  }
  if (tid == TPB - 1) {
    int n = base + incl;
    scal[0] = n;
    scal[1] = n >> 1;   // n // 2, n >= 0
  }
}

// Reduce 3136 block partials -> scalar, scale by LAMBDA_COORD.
__global__ void yolo_final(const float* __restrict__ partials, float* __restrict__ out) {
  int tid  = threadIdx.x;
  int lane = tid & 31, wave = tid >> 5;
  float s = 0.0f;
  for (int i = tid; i < NBLK; i += TPB) s += partials[i];
  float wtot = wave_reduce_wmma(s);
  __shared__ float fsum[8];
  if (lane == 0) fsum[wave] = wtot;
  __syncthreads();
  if (tid == 0) {
    float tot = 0.0f;
    for (int i = 0; i < 8; ++i) tot += fsum[i];
    out[0] = 5.0f * tot;
  }
}

extern "C" void kernel_launch(void* const* d_in, const int* in_sizes, int n_in,
                              void* d_out, int out_size, void* d_ws, size_t ws_size,
                              hipStream_t stream) {
  (void)in_sizes; (void)n_in; (void)out_size;
  const float* preds   = (const float*)d_in[0];
  const float* targets = (const float*)d_in[1];

  int*   counts   = (int*)d_ws;
  int*   offs     = counts + NBLK;
  int*   scal     = offs + NBLK;          // [0]=n, [1]=n/2
  float* partials = (float*)(scal + 2);
  float* cpack    = partials + NBLK;      // fast path only

  size_t need_fast = ((size_t)NBLK * 3 + 2 + (size_t)NROWS) * sizeof(float);

  if (ws_size >= need_fast) {
    // Fast path: big arrays streamed exactly once (192 MB), then ~3.2 MB repass.
    yolo_fuse <<<NBLK, TPB, 0, stream>>>(preds, targets, cpack, counts);
    yolo_scan <<<1,    TPB, 0, stream>>>(counts, offs, scal);
    yolo_wsum <<<NBLK, TPB, 0, stream>>>(cpack, counts, offs, scal, partials);
    yolo_final<<<1,    TPB, 0, stream>>>(partials, (float*)d_out);
  } else {
    // Fallback: mask pre-pass + full recompute pass.
    yolo_count<<<NBLK, TPB, 0, stream>>>(targets, counts);
    yolo_scan <<<1,    TPB, 0, stream>>>(counts, offs, scal);
    yolo_main <<<NBLK, TPB, 0, stream>>>(preds, targets, offs, scal, partials);
    yolo_final<<<1,    TPB, 0, stream>>>(partials, (float*)d_out);
  }
}